// ModifiedMultiheadAttention_63848983822744
// MI455X (gfx1250) — compile-verified
//
#include <hip/hip_runtime.h>

// CDNA5 / gfx1250, wave32. fp32 WMMA path: V_WMMA_F32_16X16X4_F32.
typedef __attribute__((ext_vector_type(2))) float v2f;
typedef __attribute__((ext_vector_type(8))) float v8f;
typedef __attribute__((ext_vector_type(4))) unsigned int u32x4;
typedef __attribute__((ext_vector_type(8))) int i32x8;
typedef __attribute__((ext_vector_type(4))) int i32x4;

#define WMMA_F32(a, b, c) \
  __builtin_amdgcn_wmma_f32_16x16x4_f32(false, (a), false, (b), (short)0, (c), false, false)

// Problem constants (from reference)
#define BB 2
#define SS 2048
#define EE 1024
#define HH 16
#define HD 64
#define F3 (3 * EE)   // 3072

// ---------------------------------------------------------------------------
// Kernel 1: qkv[b*S+s, f] = sum_e x[b*S+s, e] * W[f, e]
// M=4096, K=1024, N=3072.  One wave = 32x64 tile (2x4 accumulators).
// ---------------------------------------------------------------------------
__global__ __launch_bounds__(256) void qkv_gemm(const float* __restrict__ X,
                                                const float* __restrict__ W,
                                                float* __restrict__ QKV) {
  const int lane = threadIdx.x & 31;
  const int wave = threadIdx.x >> 5;
  const int gwave = blockIdx.x * 8 + wave;

  const int M_TILES = (BB * SS) / 32;       // 128
  const int mt = gwave % M_TILES;
  const int nt = gwave / M_TILES;           // 0..47
  const int m0 = mt * 32;
  const int n0 = nt * 64;

  const int lm  = lane & 15;                // A: M row / B: N col (per ISA layout)
  const int lhi = (lane >> 4) * 2;          // K sub-offset within 16x4 / 4x16 frag

  v8f acc[2][4] = {};

  for (int k0 = 0; k0 < EE; k0 += 4) {
    v2f a[2], b[4];
#pragma unroll
    for (int i = 0; i < 2; ++i) {
      const float* p = X + (size_t)(m0 + i * 16 + lm) * EE + k0 + lhi;
      a[i] = *(const v2f*)p;                // {A[m][k], A[m][k+1]}
    }
#pragma unroll
    for (int j = 0; j < 4; ++j) {
      // B[k][n] = W[n][k] : lane = n, consecutive k -> contiguous in W row
      const float* p = W + (size_t)(n0 + j * 16 + lm) * EE + k0 + lhi;
      b[j] = *(const v2f*)p;
    }
#pragma unroll
    for (int i = 0; i < 2; ++i)
#pragma unroll
      for (int j = 0; j < 4; ++j)
        acc[i][j] = WMMA_F32(a[i], b[j], acc[i][j]);
  }

  // D layout: vgpr r, lane: row = r + 8*lane[4], col = lane&15
  const int chi = (lane >> 4) * 8;
#pragma unroll
  for (int i = 0; i < 2; ++i)
#pragma unroll
    for (int j = 0; j < 4; ++j)
#pragma unroll
      for (int r = 0; r < 8; ++r) {
        const int row = m0 + i * 16 + r + chi;
        const int col = n0 + j * 16 + lm;
        QKV[(size_t)row * F3 + col] = acc[i][j][r];
      }
}

// ---------------------------------------------------------------------------
// TDM: DMA a 16-row x 64-float 2D tile (row stride 3072 floats) into LDS,
// with per-row LDS padding (pad_interval code 5 == 64 DWORDs).
//   padCode: pad DWORDs - 1  (3 -> +4 DW => stride 68, 7 -> +8 DW => stride 72)
// Descriptor packing per CDNA5 ISA section 8.3/8.4 (D# groups 0 and 1).
// ---------------------------------------------------------------------------
#define HAVE_TDM __has_builtin(__builtin_amdgcn_tensor_load_to_lds)

#if HAVE_TDM
__device__ __forceinline__ void tdm_load_tile(const float* gsrc, unsigned ldsByteAddr,
                                              int padCode) {
  const unsigned long long ga = (unsigned long long)(size_t)gsrc;
  u32x4 g0;
  g0[0] = 1u;                                           // count=1 valid descriptor
  g0[1] = ldsByteAddr;                                  // lds_addr (bytes)
  g0[2] = (unsigned)(ga & 0xffffffffu);                 // global_addr[31:0]
  g0[3] = (unsigned)((ga >> 32) & 0x01ffffffu)          // global_addr[56:32]
        | (2u << 30);                                   // type = 2 ("image")
  i32x8 g1;
  g1[0] = (2 << 16)                                     // data_size = 4B
        | (1 << 20)                                     // pad_enable
        | (5 << 22)                                     // pad_interval: 64 DWORDs
        | (padCode << 25);                              // pad_amount
  g1[1] = (HD & 0xffff) << 16;                          // tensor_dim0 = 64 (lo16)
  g1[2] = ((HD >> 16) & 0xffff) | (16 << 16);           // dim0 hi16 | tensor_dim1=16 lo16
  g1[3] = 0 | (HD << 16);                               // dim1 hi16 | tile_dim0 = 64
  g1[4] = 16 | (0 << 16);                               // tile_dim1 = 16 | tile_dim2 = 0
  g1[5] = F3;                                           // tensor_dim0_stride = 3072 (lo32)
  g1[6] = 0;                                            // stride hi16 | dim1_stride lo16
  g1[7] = 0;
  const i32x4 z4 = {0, 0, 0, 0};                        // groups 2/3: 2D tensor
  const i32x8 z8 = {0, 0, 0, 0, 0, 0, 0, 0};
  __builtin_amdgcn_tensor_load_to_lds(g0, g1, z4, z4, z8, 0);
}
#endif

// ---------------------------------------------------------------------------
// Kernel 2: fused attention (no softmax):
//   O[s,:] = sum_t ( (q[s]·k[t]) / 8 ) * v[t,:]   per (b,h)
// Block = 8 waves -> 128 q rows; K/V streamed in 16-row tiles via TDM,
// double-buffered in LDS.
// ---------------------------------------------------------------------------
#define KSTR 68   // LDS row stride (floats) for K tile -- conflict-free B reads
#define VSTR 72   // LDS row stride (floats) for V tile
#define SSTR 20   // LDS row stride for per-wave 16x16 score tile

__global__ __launch_bounds__(256) void attn_fused(const float* __restrict__ QKV,
                                                  float* __restrict__ OUT) {
  __shared__ float Kbuf[2][16 * KSTR];
  __shared__ float Vbuf[2][16 * VSTR];
  __shared__ float Sbuf[8][16 * SSTR];

  const int lane = threadIdx.x & 31;
  const int wave = threadIdx.x >> 5;

  int tmp = blockIdx.x;
  const int qt = tmp & 15; tmp >>= 4;       // S/128 = 16 q-tiles
  const int h  = tmp & 15; tmp >>= 4;       // H = 16
  const int b  = tmp;                       // B = 2
  const int q0 = qt * 128 + wave * 16;

  const int lm  = lane & 15;
  const int lhi = (lane >> 4) * 2;

  const size_t headBase = (size_t)(b * SS) * F3 + (size_t)h * (3 * HD);

  // Preload this wave's Q fragments (16 rows x 64 d) into registers.
  v2f aQ[16];
  {
    const float* qrow = QKV + headBase + (size_t)(q0 + lm) * F3;  // q at offset 0
#pragma unroll
    for (int kk = 0; kk < 16; ++kk)
      aQ[kk] = *(const v2f*)(qrow + kk * 4 + lhi);
  }

#if HAVE_TDM
  // Flat shared-pointer low 32 bits == LDS byte offset (LDS aperture mapping).
  const unsigned kLds0 = (unsigned)(size_t)&Kbuf[0][0];
  const unsigned kLds1 = (unsigned)(size_t)&Kbuf[1][0];
  const unsigned vLds0 = (unsigned)(size_t)&Vbuf[0][0];
  const unsigned vLds1 = (unsigned)(size_t)&Vbuf[1][0];

  // Prologue: wave 0 kicks off tile 0 into buffer 0.
  if (wave == 0) {
    tdm_load_tile(QKV + headBase + HD,     kLds0, 3);   // K rows, +4 DW pad
    tdm_load_tile(QKV + headBase + 2 * HD, vLds0, 7);   // V rows, +8 DW pad
  }
#endif

  v8f oAcc[4] = {};
  int p = 0;

  for (int t0 = 0; t0 < SS; t0 += 16) {
#if HAVE_TDM
    if (wave == 0) __builtin_amdgcn_s_wait_tensorcnt(0);  // tile t0 landed
    __syncthreads();                                      // release all waves
    if (wave == 0 && (t0 + 16) < SS) {                    // prefetch tile t0+16
      const float* base = QKV + headBase + (size_t)(t0 + 16) * F3;
      tdm_load_tile(base + HD,     p ? kLds0 : kLds1, 3);
      tdm_load_tile(base + 2 * HD, p ? vLds0 : vLds1, 7);
    }
#else
    {  // fallback: cooperative float4 loads
      const int r = threadIdx.x >> 4;
      const int c = (threadIdx.x & 15) * 4;
      const float* base = QKV + headBase + (size_t)(t0 + r) * F3;
      *(float4*)&Kbuf[p][r * KSTR + c] = *(const float4*)(base + HD + c);
      *(float4*)&Vbuf[p][r * VSTR + c] = *(const float4*)(base + 2 * HD + c);
    }
    __syncthreads();
#endif

    // ---- S(16x16) = Q(16x64) * K^T(64x16) ----
    v8f sAcc = {};
#pragma unroll
    for (int kk = 0; kk < 16; ++kk) {
      // B[k][n] = K[t0+n][k]: lane = n (t row), reads 2 consecutive d
      v2f bK = *(const v2f*)&Kbuf[p][lm * KSTR + kk * 4 + lhi];
      sAcc = WMMA_F32(aQ[kk], bK, sAcc);
    }

    // scale by 1/sqrt(64) and park in LDS (D layout -> row major)
#pragma unroll
    for (int r = 0; r < 8; ++r)
      Sbuf[wave][(r + ((lane >> 4) * 8)) * SSTR + lm] = sAcc[r] * 0.125f;

    // ---- O(16x64) += S(16x16) * V(16x64) ----
#pragma unroll
    for (int ks = 0; ks < 4; ++ks) {
      v2f aS = *(const v2f*)&Sbuf[wave][lm * SSTR + ks * 4 + lhi];
      const int kr = ks * 4 + lhi;
#pragma unroll
      for (int ntile = 0; ntile < 4; ++ntile) {
        v2f bV;
        bV.x = Vbuf[p][kr * VSTR + ntile * 16 + lm];
        bV.y = Vbuf[p][(kr + 1) * VSTR + ntile * 16 + lm];
        oAcc[ntile] = WMMA_F32(aS, bV, oAcc[ntile]);
      }
    }
#if !HAVE_TDM
    __syncthreads();   // protect K/V tiles before next iteration overwrites
#endif
    p ^= 1;
  }

  // ---- store: o[b, s, h*64 + d] ----
#pragma unroll
  for (int ntile = 0; ntile < 4; ++ntile)
#pragma unroll
    for (int r = 0; r < 8; ++r) {
      const int s = q0 + r + ((lane >> 4) * 8);
      const int d = h * HD + ntile * 16 + lm;
      OUT[(size_t)(b * SS + s) * EE + d] = oAcc[ntile][r];
    }
}

// ---------------------------------------------------------------------------
extern "C" void kernel_launch(void* const* d_in, const int* in_sizes, int n_in,
                              void* d_out, int out_size, void* d_ws, size_t ws_size,
                              hipStream_t stream) {
  (void)in_sizes; (void)n_in; (void)out_size; (void)ws_size;
  const float* x = (const float*)d_in[0];     // [B,S,E]
  const float* w = (const float*)d_in[1];     // [3E,E]
  float* out = (float*)d_out;                 // [B,S,E]
  float* qkv = (float*)d_ws;                  // [B*S, 3E] = 50.3 MB scratch

  // GEMM1: 128*48 = 6144 waves, 8 waves/block
  qkv_gemm<<<768, 256, 0, stream>>>(x, w, qkv);
  // Attention: B*H*(S/128) = 512 blocks
  attn_fused<<<512, 256, 0, stream>>>(qkv, out);
}